// PVConv_48404281426404
// MI455X (gfx1250) — compile-verified
//
#include <hip/hip_runtime.h>
#include <hip/hip_bf16.h>

typedef __attribute__((ext_vector_type(2))) float v2f;
typedef __attribute__((ext_vector_type(8))) float v8f;

#define RR 32
#define BB 8
#define CC 64
#define NN 8192
#define VOX 32768            // 32^3
#define VOXTOT (BB*CC*VOX)   // 16,777,216
#define PTOT   (BB*CC*NN)    // 4,194,304

// ---------------- utility ----------------
__global__ void zero_kernel(float* __restrict__ p, int n) {
    int i = blockIdx.x * 256 + threadIdx.x;
    if (i < n) p[i] = 0.f;
}

// ---------------- coord stats: per-batch mean + 2*max||c-mean|| ----------------
__global__ void coord_stats(const float* __restrict__ coords, float* __restrict__ stats) {
    int b = blockIdx.x, tid = threadIdx.x;
    const float* cb = coords + (size_t)b * 3 * NN;
    __shared__ float r0[256], r1[256], r2[256];
    float sx = 0.f, sy = 0.f, sz = 0.f;
    for (int n = tid; n < NN; n += 256) {
        sx += cb[n]; sy += cb[NN + n]; sz += cb[2 * NN + n];
    }
    r0[tid] = sx; r1[tid] = sy; r2[tid] = sz; __syncthreads();
    for (int s = 128; s > 0; s >>= 1) {
        if (tid < s) { r0[tid] += r0[tid + s]; r1[tid] += r1[tid + s]; r2[tid] += r2[tid + s]; }
        __syncthreads();
    }
    __shared__ float mxs, mys, mzs;
    if (tid == 0) { mxs = r0[0] / (float)NN; mys = r1[0] / (float)NN; mzs = r2[0] / (float)NN; }
    __syncthreads();
    float mX = mxs, mY = mys, mZ = mzs;
    float mn = 0.f;
    for (int n = tid; n < NN; n += 256) {
        float dx = cb[n] - mX, dy = cb[NN + n] - mY, dz = cb[2 * NN + n] - mZ;
        mn = fmaxf(mn, sqrtf(dx * dx + dy * dy + dz * dz));
    }
    __syncthreads();
    r0[tid] = mn; __syncthreads();
    for (int s = 128; s > 0; s >>= 1) {
        if (tid < s) r0[tid] = fmaxf(r0[tid], r0[tid + s]);
        __syncthreads();
    }
    if (tid == 0) {
        stats[b * 4 + 0] = mX; stats[b * 4 + 1] = mY; stats[b * 4 + 2] = mZ;
        stats[b * 4 + 3] = r0[0] * 2.0f;
    }
}

// ---------------- scatter-add features into voxel grid ----------------
__global__ void scatter_kernel(const float* __restrict__ coords, const float* __restrict__ feats,
                               const float* __restrict__ stats, float* __restrict__ vox_sum,
                               float* __restrict__ cnt, float* __restrict__ ncrd) {
    int gid = blockIdx.x * 256 + threadIdx.x;           // B*N threads
    if (gid >= BB * NN) return;
    int b = gid / NN, n = gid % NN;
    const float* cb = coords + (size_t)b * 3 * NN;
    float mX = stats[b * 4 + 0], mY = stats[b * 4 + 1], mZ = stats[b * 4 + 2];
    float denom = stats[b * 4 + 3];
    float ncx = ((cb[n]          - mX) / denom + 0.5f) * (float)RR;
    float ncy = ((cb[NN + n]     - mY) / denom + 0.5f) * (float)RR;
    float ncz = ((cb[2 * NN + n] - mZ) / denom + 0.5f) * (float)RR;
    ncx = fminf(fmaxf(ncx, 0.f), (float)(RR - 1));
    ncy = fminf(fmaxf(ncy, 0.f), (float)(RR - 1));
    ncz = fminf(fmaxf(ncz, 0.f), (float)(RR - 1));
    float* nb = ncrd + (size_t)b * 3 * NN;
    nb[n] = ncx; nb[NN + n] = ncy; nb[2 * NN + n] = ncz;
    int vx = __float2int_rn(ncx), vy = __float2int_rn(ncy), vz = __float2int_rn(ncz);
    int idx = (vx * RR + vy) * RR + vz;
    atomicAdd(&cnt[b * VOX + idx], 1.0f);
    const float* fb = feats + (size_t)b * CC * NN + n;
    float* vb = vox_sum + (size_t)b * CC * VOX + idx;
    for (int c = 0; c < CC; ++c) atomicAdd(&vb[c * VOX], fb[c * NN]);
}

// ---------------- divide by counts (in place) ----------------
__global__ void finalize_vox(float* __restrict__ vox, const float* __restrict__ cnt) {
    int i = blockIdx.x * 256 + threadIdx.x;
    if (i >= VOXTOT) return;
    int s = i & (VOX - 1);
    int b = i >> 21;                                    // / (64*32768)
    vox[i] = vox[i] / fmaxf(cnt[(b << 15) + s], 1.0f);
}

// ---------------- 3x3x3 conv 64->64 via implicit GEMM, f32 WMMA 16x16x4 ----------------
// grid = B*R*R blocks (one per (b,x,y) row of 32 z), block = 256 = 8 wave32
// wave w: co0 = (w&3)*16, z0 = (w>>2)*16 ; K = ci*27 taps, grouped (txy,dz,ci4)
__global__ __launch_bounds__(256) void conv3d_wmma(const float* __restrict__ X,
                                                   const float* __restrict__ W,
                                                   const float* __restrict__ bias,
                                                   float* __restrict__ Y) {
    int blk = blockIdx.x;
    int b = blk / (RR * RR);
    int xy = blk % (RR * RR);
    int xx = xy / RR, yy = xy % RR;
    int tid = threadIdx.x;
    int wave = tid >> 5, lane = tid & 31;
    int co0 = (wave & 3) * 16;
    int z0  = (wave >> 2) * 16;
    int l = lane & 15;
    int lhalf = lane >> 4;                              // K sub-select

    __shared__ float Xs[CC][36];                        // ci x (z in -1..32), padded
    __shared__ float Ws[192][65];                       // k_local = dz*64+ci, co (padded)

    v8f acc = {0.f, 0.f, 0.f, 0.f, 0.f, 0.f, 0.f, 0.f};

    for (int txy = 0; txy < 9; ++txy) {
        int dx = txy / 3 - 1, dy = txy % 3 - 1;
        int xs = xx + dx, ys = yy + dy;
        bool inXY = (xs >= 0) && (xs < RR) && (ys >= 0) && (ys < RR);
        const float* xb = X + (size_t)b * CC * VOX + (xs * RR + ys) * RR;
        __syncthreads();
        // stage input row: 64 ci x 34 z
        for (int i = tid; i < CC * 34; i += 256) {
            int ci = i / 34, zi = i % 34;
            int z = zi - 1;
            float v = 0.f;
            if (inXY && z >= 0 && z < RR) v = xb[ci * VOX + z];
            Xs[ci][zi] = v;
        }
        // stage weights: k_local = dz*64+ci over all 64 co
        for (int i = tid; i < 192 * 64; i += 256) {
            int k = i >> 6, co = i & 63;
            int dz = k >> 6, ci = k & 63;
            Ws[k][co] = W[(co * CC + ci) * 27 + txy * 3 + dz];
        }
        __syncthreads();
        #pragma unroll 4
        for (int kg = 0; kg < 48; ++kg) {
            int k0 = kg * 4;
            int dz = k0 >> 6;
            int cA = (k0 & 63) + lhalf * 2;             // A: K = k0 + {0,1}/{2,3}
            int zi = z0 + l + dz;                       // Xs index for z' = z+dz-1
            v2f a, bm;
            a.x  = Xs[cA][zi];
            a.y  = Xs[cA + 1][zi];
            bm.x = Ws[k0 + lhalf * 2][co0 + l];
            bm.y = Ws[k0 + lhalf * 2 + 1][co0 + l];
            acc = __builtin_amdgcn_wmma_f32_16x16x4_f32(
                false, a, false, bm, (short)0, acc, false, false);
        }
    }
    // D layout: vgpr r -> M = r + 8*lhalf (z offset), N = l (co offset)
    int co = co0 + l;
    float bv = bias[co];
    float* yb = Y + (size_t)(b * CC + co) * VOX + (xx * RR + yy) * RR + z0 + lhalf * 8;
    #pragma unroll
    for (int r = 0; r < 8; ++r) yb[r] = acc[r] + bv;
}

// ---------------- BN train stats -> scale/shift per channel ----------------
__global__ void bn_stats(const float* __restrict__ X, const float* __restrict__ g,
                         const float* __restrict__ bt, float* __restrict__ ss,
                         int spatial) {
    int c = blockIdx.x, tid = threadIdx.x;
    long total = (long)BB * spatial;
    double s = 0.0, s2 = 0.0;
    for (long j = tid; j < total; j += 256) {
        long b = j / spatial, sp = j % spatial;
        float v = X[(b * CC + c) * spatial + sp];
        s += v; s2 += (double)v * v;
    }
    __shared__ double rs[256], rs2[256];
    rs[tid] = s; rs2[tid] = s2; __syncthreads();
    for (int st = 128; st > 0; st >>= 1) {
        if (tid < st) { rs[tid] += rs[tid + st]; rs2[tid] += rs2[tid + st]; }
        __syncthreads();
    }
    if (tid == 0) {
        double mean = rs[0] / (double)total;
        double var = rs2[0] / (double)total - mean * mean;
        float scale = g[c] * rsqrtf((float)var + 1e-4f);
        ss[c] = scale;
        ss[CC + c] = bt[c] - (float)mean * scale;
    }
}

// ---------------- apply BN + LeakyReLU ----------------
__global__ void bn_apply_leaky(const float* __restrict__ X, float* __restrict__ Y,
                               const float* __restrict__ ss, int spatial, int total) {
    int i = blockIdx.x * 256 + threadIdx.x;
    if (i >= total) return;
    int c = (i / spatial) & (CC - 1);
    float v = X[i] * ss[c] + ss[CC + c];
    Y[i] = v >= 0.f ? v : 0.1f * v;
}

// ---------------- point branch: 64x64 shared MLP (raw, pre-BN) ----------------
__global__ void point_mlp(const float* __restrict__ feats, const float* __restrict__ Wm,
                          const float* __restrict__ bm, float* __restrict__ p) {
    int gid = blockIdx.x * 256 + threadIdx.x;           // B*C*N
    if (gid >= PTOT) return;
    int n = gid & (NN - 1);
    int t = gid >> 13;
    int co = t & (CC - 1);
    int b = t >> 6;
    const float* f = feats + (size_t)b * CC * NN + n;
    const float* w = Wm + co * CC;
    float s = bm[co];
    #pragma unroll 8
    for (int ci = 0; ci < CC; ++ci) s += w[ci] * f[ci * NN];
    p[gid] = s;
}

// ---------------- trilinear devoxelize + BN3+leaky point branch + fuse ----------------
__global__ void fuse_kernel(const float* __restrict__ v, const float* __restrict__ nc,
                            const float* __restrict__ p, const float* __restrict__ ss3,
                            float* __restrict__ out) {
    int gid = blockIdx.x * 256 + threadIdx.x;           // B*C*N
    if (gid >= PTOT) return;
    int n = gid & (NN - 1);
    int t = gid >> 13;
    int co = t & (CC - 1);
    int b = t >> 6;
    const float* nb = nc + (size_t)b * 3 * NN;
    float x = nb[n], y = nb[NN + n], z = nb[2 * NN + n];
    int x0 = (int)floorf(x), y0 = (int)floorf(y), z0 = (int)floorf(z);
    int x1 = min(x0 + 1, RR - 1), y1 = min(y0 + 1, RR - 1), z1 = min(z0 + 1, RR - 1);
    float fx = x - (float)x0, fy = y - (float)y0, fz = z - (float)z0;
    float gx0 = 1.f - fx, gy0 = 1.f - fy, gz0 = 1.f - fz;
    const float* vb = v + (size_t)(b * CC + co) * VOX;
    float acc = 0.f;
    acc += vb[(x0 * RR + y0) * RR + z0] * (gx0 * gy0 * gz0);
    acc += vb[(x0 * RR + y0) * RR + z1] * (gx0 * gy0 * fz);
    acc += vb[(x0 * RR + y1) * RR + z0] * (gx0 * fy * gz0);
    acc += vb[(x0 * RR + y1) * RR + z1] * (gx0 * fy * fz);
    acc += vb[(x1 * RR + y0) * RR + z0] * (fx * gy0 * gz0);
    acc += vb[(x1 * RR + y0) * RR + z1] * (fx * gy0 * fz);
    acc += vb[(x1 * RR + y1) * RR + z0] * (fx * fy * gz0);
    acc += vb[(x1 * RR + y1) * RR + z1] * (fx * fy * fz);
    float pv = p[gid] * ss3[co] + ss3[CC + co];
    pv = pv >= 0.f ? pv : 0.1f * pv;
    out[gid] = acc + pv;
}

extern "C" void kernel_launch(void* const* d_in, const int* in_sizes, int n_in,
                              void* d_out, int out_size, void* d_ws, size_t ws_size,
                              hipStream_t stream) {
    (void)in_sizes; (void)n_in; (void)out_size; (void)ws_size;
    const float* features = (const float*)d_in[0];
    const float* coords   = (const float*)d_in[1];
    const float* c1w = (const float*)d_in[2];
    const float* c1b = (const float*)d_in[3];
    const float* g1  = (const float*)d_in[4];
    const float* b1  = (const float*)d_in[5];
    const float* c2w = (const float*)d_in[6];
    const float* c2b = (const float*)d_in[7];
    const float* g2  = (const float*)d_in[8];
    const float* b2  = (const float*)d_in[9];
    const float* mw  = (const float*)d_in[10];
    const float* mb  = (const float*)d_in[11];
    const float* g3  = (const float*)d_in[12];
    const float* b3  = (const float*)d_in[13];

    float* wsf    = (float*)d_ws;
    float* buf0   = wsf;                        // 16,777,216 (vox sum -> vox feat -> conv2 raw)
    float* buf1   = buf0 + VOXTOT;              // 16,777,216 (conv1 raw -> v1)
    float* cnt    = buf1 + VOXTOT;              // 262,144
    float* ncrd   = cnt + BB * VOX;             // 196,608
    float* pbuf   = ncrd + BB * 3 * NN;         // 4,194,304
    float* cstats = pbuf + PTOT;                // 32
    float* ss     = cstats + 32;                // 128 (scale/shift, reused)
    float* ss3    = ss + 128;                   // 128 (point-branch scale/shift)

    float* out_fused  = (float*)d_out;          // [B,C,N]
    float* out_coords = out_fused + PTOT;       // [B,3,N]
    float* out_v      = out_coords + BB * 3 * NN; // [B,C,32,32,32]

    // 1) zero voxel accumulators
    zero_kernel<<<(VOXTOT + 255) / 256, 256, 0, stream>>>(buf0, VOXTOT);
    zero_kernel<<<(BB * VOX + 255) / 256, 256, 0, stream>>>(cnt, BB * VOX);
    // 2) voxelize
    coord_stats<<<BB, 256, 0, stream>>>(coords, cstats);
    scatter_kernel<<<(BB * NN + 255) / 256, 256, 0, stream>>>(coords, features, cstats,
                                                              buf0, cnt, ncrd);
    finalize_vox<<<(VOXTOT + 255) / 256, 256, 0, stream>>>(buf0, cnt);
    // 3) conv1 -> BN1 -> leaky
    conv3d_wmma<<<BB * RR * RR, 256, 0, stream>>>(buf0, c1w, c1b, buf1);
    bn_stats<<<CC, 256, 0, stream>>>(buf1, g1, b1, ss, VOX);
    bn_apply_leaky<<<(VOXTOT + 255) / 256, 256, 0, stream>>>(buf1, buf1, ss, VOX, VOXTOT);
    // 4) conv2 -> BN2 -> leaky (writes v output directly)
    conv3d_wmma<<<BB * RR * RR, 256, 0, stream>>>(buf1, c2w, c2b, buf0);
    bn_stats<<<CC, 256, 0, stream>>>(buf0, g2, b2, ss, VOX);
    bn_apply_leaky<<<(VOXTOT + 255) / 256, 256, 0, stream>>>(buf0, out_v, ss, VOX, VOXTOT);
    // 5) coords passthrough
    hipMemcpyAsync(out_coords, coords, (size_t)BB * 3 * NN * sizeof(float),
                   hipMemcpyDeviceToDevice, stream);
    // 6) point branch + fuse
    point_mlp<<<(PTOT + 255) / 256, 256, 0, stream>>>(features, mw, mb, pbuf);
    bn_stats<<<CC, 256, 0, stream>>>(pbuf, g3, b3, ss3, NN);
    fuse_kernel<<<(PTOT + 255) / 256, 256, 0, stream>>>(out_v, ncrd, pbuf, ss3, out_fused);
}